// MambaBlock_69947837383060
// MI455X (gfx1250) — compile-verified
//
#include <hip/hip_runtime.h>
#include <hip/hip_bf16.h>

// ---------------------------------------------------------------------------
// Mamba block for MI455X (gfx1250, wave32).
//  - GEMMs: V_WMMA_F32_16X16X4_F32 (fp32 A/B, fp32 acc) for full fidelity.
//  - Selective scan: chunked through LDS using the Tensor Data Mover
//    (tensor_load_to_lds) for the shared B|C tile and async global->LDS
//    (global_load_async_to_lds_b128) for the per-channel delta / x tiles.
// ---------------------------------------------------------------------------

#define SEQ_L   2048
#define D_MODEL 512
#define D_INNER 1024
#define N_STATE 16
#define DT_RANK 32

typedef __attribute__((ext_vector_type(2))) float v2f;
typedef __attribute__((ext_vector_type(8))) float v8f;
typedef __attribute__((ext_vector_type(4))) unsigned int u32x4;
typedef __attribute__((ext_vector_type(8))) unsigned int u32x8;

enum { EPI_NONE = 0, EPI_BIAS_SOFTPLUS = 1 };

// ---------------------------------------------------------------------------
// Generic GEMM:  C[M,N] = A[M,K] (row-major, lda) * W[N,K]^T (row-major, ldb)
// One wave computes one 16x16 C tile. fp32 WMMA, K in steps of 4.
//
// fp32 A-matrix 16x4 VGPR layout (ISA 7.12.2): lane = 16*half + r,
//   a[j] = A[M=r, K=2*half + j].  B 4x16 is symmetric: b[j] = B[K=2*half+j, N=r],
//   and since B[k,n] = W[n,k], lane r reads row (tile_n*16+r) of W.
// C/D layout: VGPR j, lanes 0-15 -> M=j, lanes 16-31 -> M=j+8; N = r.
// ---------------------------------------------------------------------------
template <int EPI>
__global__ __launch_bounds__(256) void wmma_gemm_nt(
    const float* __restrict__ A, int lda,
    const float* __restrict__ W, int ldb,
    float* __restrict__ C, int ldc,
    int M, int N, int K,
    const float* __restrict__ bias)
{
    const int tilesN = N >> 4;
    const int totalTiles = (M >> 4) * tilesN;
    const int wave = blockIdx.x * (blockDim.x >> 5) + (threadIdx.x >> 5);
    if (wave >= totalTiles) return;            // wave-uniform: EXEC stays all-1s

    const int tm   = wave / tilesN;
    const int tn   = wave % tilesN;
    const int lane = threadIdx.x & 31;
    const int half = lane >> 4;
    const int r    = lane & 15;

    const float* ap = A + (size_t)(tm * 16 + r) * lda + 2 * half;
    const float* bp = W + (size_t)(tn * 16 + r) * ldb + 2 * half;

    v8f acc = {};
    for (int k0 = 0; k0 < K; k0 += 4) {
        // prefetch 256 floats (1KB) ahead within this row of A / W
        if (((k0 & 63) == 0) && (k0 + 260 <= K)) {
            __builtin_prefetch(ap + 256, 0, 1);   // -> global_prefetch_b8
            __builtin_prefetch(bp + 256, 0, 1);
        }
        v2f a = *(const v2f*)ap;                  // 8B load, aligned (k0,lda even)
        v2f b = *(const v2f*)bp;
        acc = __builtin_amdgcn_wmma_f32_16x16x4_f32(
            /*neg_a=*/false, a, /*neg_b=*/false, b,
            /*c_mod=*/(short)0, acc, /*reuse_a=*/false, /*reuse_b=*/false);
        ap += 4;
        bp += 4;
    }

    const int col = tn * 16 + r;
    float* crow = C + (size_t)(tm * 16 + 8 * half) * ldc + col;
#pragma unroll
    for (int j = 0; j < 8; ++j) {
        float v = acc[j];
        if (EPI == EPI_BIAS_SOFTPLUS) {
            v += bias[col];
            v = (v > 20.f) ? v : log1pf(expf(v));   // softplus
        }
        crow[(size_t)j * ldc] = v;
    }
}

// ---------------------------------------------------------------------------
// Depthwise causal conv1d (width 4, cross-correlation, left pad 3) + bias + SiLU.
// Input xz is the in_proj output (L x 2*D_INNER); xp = columns [0, D_INNER).
// ---------------------------------------------------------------------------
__global__ __launch_bounds__(256) void conv_silu_kernel(
    const float* __restrict__ xz, const float* __restrict__ w,
    const float* __restrict__ b, float* __restrict__ out)
{
    const int idx = blockIdx.x * blockDim.x + threadIdx.x;   // l*D_INNER + d
    if (idx >= SEQ_L * D_INNER) return;
    const int l = idx >> 10;
    const int d = idx & (D_INNER - 1);

    float acc = b[d];
    const float* wd = w + d * 4;
#pragma unroll
    for (int j = 0; j < 4; ++j) {
        const int li = l - 3 + j;
        if (li >= 0)
            acc = fmaf(xz[(size_t)li * (2 * D_INNER) + d], wd[j], acc);
    }
    out[idx] = acc / (1.f + expf(-acc));                      // SiLU
}

// ---------------------------------------------------------------------------
// Selective scan, fused with +x*D skip and *silu(z) gate.
// Block handles 16 channels; thread t owns state (d = blk*16 + t/16, n = t%16).
// L is processed in 64-step chunks staged into LDS:
//   * B|C tile (64 x 32 f32, shared by the whole block): one TDM
//     tensor_load_to_lds issued by wave 0 (TDM ignores EXEC -> 1 DMA/block).
//   * delta / x tiles (64 x 16 f32 each): global_load_async_to_lds_b128.
// Recurrence matches the reference's parallel-cumsum form exactly:
//   h[l] = clip(exp(delta*A), 1e-6) * (h[l-1] + delta*B*x)
//   y[l,d] = sum_n h[l,d,n] * C[l,n]
// ---------------------------------------------------------------------------
__global__ __launch_bounds__(256) void scan_kernel(
    const float* __restrict__ xdbl,   // (L, 64): [delta_r | B | C]
    const float* __restrict__ delta,  // (L, D_INNER) post-softplus
    const float* __restrict__ xcs,    // (L, D_INNER) conv+silu output
    const float* __restrict__ xz,     // (L, 2*D_INNER): z = cols [D_INNER, 2*D_INNER)
    const float* __restrict__ A_log,  // (D_INNER, N_STATE)
    const float* __restrict__ Dp,     // (D_INNER)
    float* __restrict__ outpre)       // (L, D_INNER) gated output
{
    __shared__ float sBC[64 * 32];    // chunk rows of [B | C]
    __shared__ float sD [64 * 16];    // delta chunk for this block's 16 channels
    __shared__ float sX [64 * 16];    // x chunk

    const int t     = threadIdx.x;
    const int n     = t & 15;
    const int dl    = t >> 4;
    const int dbase = blockIdx.x * 16;
    const int d     = dbase + dl;

    const float Aval = -expf(A_log[d * N_STATE + n]);
    const float Dval = Dp[d];

    float h = 0.f;
    for (int l0 = 0; l0 < SEQ_L; l0 += 64) {
        // ---- TDM: 2-D tile xdbl[l0:l0+64, 32:64) -> sBC (row-major) --------
        if (t < 32) {   // wave 0 only; TDM ignores EXEC, so this is 1 DMA op
            const unsigned long long ga =
                (unsigned long long)(xdbl + (size_t)l0 * 64 + DT_RANK);
            const unsigned ldsa = (unsigned)(unsigned long long)(&sBC[0]);
            u32x4 g0;
            u32x8 g1;
            g0[0] = 1u;                                   // count=1, user mode
            g0[1] = ldsa;                                 // lds_addr
            g0[2] = (unsigned)(ga & 0xFFFFFFFFu);         // global_addr[31:0]
            g0[3] = (unsigned)((ga >> 32) & 0x1FFFFFFu)   // global_addr[56:32]
                  | (2u << 30);                           // type=2 (image)
            g1[0] = (2u << 16);                           // data_size=2 (4 bytes)
            g1[1] = (32u << 16);                          // tensor_dim0 = 32
            g1[2] = (64u << 16);                          // tensor_dim1 = 64
            g1[3] = (32u << 16);                          // tile_dim0 = 32
            g1[4] = 64u;                                  // tile_dim1 = 64
            g1[5] = 64u;                                  // tensor_dim0_stride = 64
            g1[6] = 0u;
            g1[7] = 0u;
            asm volatile("tensor_load_to_lds %0, %1"
                         :: "s"(g0), "s"(g1) : "memory");
        }

        // ---- async global->LDS: delta / x tiles (one b128 each per thread) -
        {
            const int elem = t * 4;            // 1024 floats = 64 rows x 16 ch
            const int row  = elem >> 4;
            const int col  = elem & 15;
            const unsigned ldsD = (unsigned)(unsigned long long)(&sD[elem]);
            const unsigned ldsX = (unsigned)(unsigned long long)(&sX[elem]);
            const unsigned long long gaD =
                (unsigned long long)(delta + (size_t)(l0 + row) * D_INNER + dbase + col);
            const unsigned long long gaX =
                (unsigned long long)(xcs + (size_t)(l0 + row) * D_INNER + dbase + col);
            asm volatile("global_load_async_to_lds_b128 %0, %1, off"
                         :: "v"(ldsD), "v"(gaD) : "memory");
            asm volatile("global_load_async_to_lds_b128 %0, %1, off"
                         :: "v"(ldsX), "v"(gaX) : "memory");
        }

        asm volatile("s_wait_asynccnt 0x0" ::: "memory");
        __builtin_amdgcn_s_wait_tensorcnt(0);
        __syncthreads();

        // ---- 64 serial recurrence steps entirely out of LDS ----------------
        for (int ll = 0; ll < 64; ++ll) {
            const float dlt = sD[ll * 16 + dl];
            const float xv  = sX[ll * 16 + dl];
            const float Bv  = sBC[ll * 32 + n];
            const float Cv  = sBC[ll * 32 + 16 + n];

            const float dA = fmaxf(expf(dlt * Aval), 1e-6f);
            h = dA * (h + dlt * Bv * xv);

            float p = h * Cv;
            p += __shfl_xor(p, 8, 16);
            p += __shfl_xor(p, 4, 16);
            p += __shfl_xor(p, 2, 16);
            p += __shfl_xor(p, 1, 16);

            if (n == 0) {
                const int l = l0 + ll;
                const float y  = p + xv * Dval;
                const float zv = xz[(size_t)l * (2 * D_INNER) + D_INNER + d];
                outpre[(size_t)l * D_INNER + d] = y * (zv / (1.f + expf(-zv)));
            }
        }
        __syncthreads();   // WAR: chunk fully consumed before restaging
    }
}

// ---------------------------------------------------------------------------
// Launch: in_proj -> conv+silu -> x_proj -> dt_proj(+softplus) -> scan -> out_proj
// ---------------------------------------------------------------------------
extern "C" void kernel_launch(void* const* d_in, const int* in_sizes, int n_in,
                              void* d_out, int out_size, void* d_ws, size_t ws_size,
                              hipStream_t stream)
{
    const float* x      = (const float*)d_in[0];
    const float* w_in   = (const float*)d_in[1];   // (2048, 512)
    const float* conv_w = (const float*)d_in[2];   // (1024, 1, 4)
    const float* conv_b = (const float*)d_in[3];   // (1024)
    const float* w_xp   = (const float*)d_in[4];   // (64, 1024)
    const float* w_dt   = (const float*)d_in[5];   // (1024, 32)
    const float* b_dt   = (const float*)d_in[6];   // (1024)
    const float* A_log  = (const float*)d_in[7];   // (1024, 16)
    const float* Dp     = (const float*)d_in[8];   // (1024)
    const float* w_out  = (const float*)d_in[9];   // (512, 1024)
    float* out = (float*)d_out;

    float* ws    = (float*)d_ws;
    float* xz    = ws;                     // 2048*2048  = 4194304 f
    float* xcs   = ws + 4194304;           // 2048*1024  = 2097152 f
    float* xdbl  = ws + 6291456;           // 2048*64    =  131072 f
    float* dlt   = ws + 6422528;           // 2048*1024  = 2097152 f
    float* opre  = ws + 8519680;           // 2048*1024  = 2097152 f

    // 1) in_proj: xz = x(2048x512) * w_in^T  -> (2048 x 2048)
    {
        const int tiles = (SEQ_L / 16) * ((2 * D_INNER) / 16);   // 16384
        wmma_gemm_nt<EPI_NONE><<<(tiles + 7) / 8, 256, 0, stream>>>(
            x, D_MODEL, w_in, D_MODEL, xz, 2 * D_INNER,
            SEQ_L, 2 * D_INNER, D_MODEL, nullptr);
    }

    // 2) causal depthwise conv + SiLU on xp half
    conv_silu_kernel<<<(SEQ_L * D_INNER) / 256, 256, 0, stream>>>(
        xz, conv_w, conv_b, xcs);

    // 3) x_proj: xdbl = xcs(2048x1024) * w_xp^T -> (2048 x 64)
    {
        const int tiles = (SEQ_L / 16) * (64 / 16);              // 512
        wmma_gemm_nt<EPI_NONE><<<(tiles + 7) / 8, 256, 0, stream>>>(
            xcs, D_INNER, w_xp, D_INNER, xdbl, 64,
            SEQ_L, 64, D_INNER, nullptr);
    }

    // 4) dt_proj + bias + softplus: dlt = softplus(xdbl[:, :32] * w_dt^T + b_dt)
    {
        const int tiles = (SEQ_L / 16) * (D_INNER / 16);         // 8192
        wmma_gemm_nt<EPI_BIAS_SOFTPLUS><<<(tiles + 7) / 8, 256, 0, stream>>>(
            xdbl, 64, w_dt, DT_RANK, dlt, D_INNER,
            SEQ_L, D_INNER, DT_RANK, b_dt);
    }

    // 5) selective scan + D skip + silu(z) gate -> opre (2048 x 1024)
    scan_kernel<<<D_INNER / 16, 256, 0, stream>>>(
        xdbl, dlt, xcs, xz, A_log, Dp, opre);

    // 6) out_proj: out = opre(2048x1024) * w_out^T -> (2048 x 512)
    {
        const int tiles = (SEQ_L / 16) * (D_MODEL / 16);         // 4096
        wmma_gemm_nt<EPI_NONE><<<(tiles + 7) / 8, 256, 0, stream>>>(
            opre, D_INNER, w_out, D_INNER, out, D_MODEL,
            SEQ_L, D_MODEL, D_INNER, nullptr);
    }
}